// Split_gcn_71957882077445
// MI455X (gfx1250) — compile-verified
//
#include <hip/hip_runtime.h>
#include <hip/hip_bf16.h>

// Shapes
#define NB   128   // batch
#define CIN  64
#define TT   128
#define VV   25
#define RR   8
#define OO   64
#define PP   (TT * VV)        // 3200 positions per sample per channel
#define EPSV 1e-5f

typedef __attribute__((ext_vector_type(2))) float v2f;
typedef __attribute__((ext_vector_type(8))) float v8f;
typedef __attribute__((ext_vector_type(4))) int   v4i;

typedef __attribute__((address_space(1))) v4i* gv4i_p;  // global int4*
typedef __attribute__((address_space(3))) v4i* lv4i_p;  // LDS int4*

#ifndef __has_builtin
#define __has_builtin(x) 0
#endif

#if __has_builtin(__builtin_amdgcn_global_load_async_to_lds_b128)
#define USE_ASYNC_LDS 1
#else
#define USE_ASYNC_LDS 0
#endif

// D = A(16x4 f32) * B(4x16 f32) + C(16x16 f32), wave32 WMMA
__device__ __forceinline__ v8f wmma_f32(v2f a, v2f b, v8f c) {
  return __builtin_amdgcn_wmma_f32_16x16x4_f32(
      /*neg_a=*/false, a, /*neg_b=*/false, b,
      /*c_mod=*/(short)0, c, /*reuse_a=*/false, /*reuse_b=*/false);
}

__device__ __forceinline__ void wait_asynccnt0() {
#if __has_builtin(__builtin_amdgcn_s_wait_asynccnt)
  __builtin_amdgcn_s_wait_asynccnt(0);
#else
  asm volatile("s_wait_asynccnt 0x0" ::: "memory");
#endif
}

// ---------------------------------------------------------------------------
// Kernel 1: fused conv1+conv2 (as one 16-row WMMA GEMM) + GroupNorm stats +
// temporal mean  ->  x1[N,R,V], x2[N,R,V]
// grid = N, block = 256 (8 waves). Reads x exactly once.
// ---------------------------------------------------------------------------
__global__ __launch_bounds__(256) void k_stats(
    const float* __restrict__ x,
    const float* __restrict__ w1, const float* __restrict__ b1,
    const float* __restrict__ g1, const float* __restrict__ be1,
    const float* __restrict__ w2, const float* __restrict__ b2,
    const float* __restrict__ g2, const float* __restrict__ be2,
    float* __restrict__ x1o, float* __restrict__ x2o) {
  const int n    = blockIdx.x;
  const int tid  = threadIdx.x;
  const int wave = tid >> 5;
  const int lane = tid & 31;
  const int lo   = lane & 15;   // M / N index within fragment
  const int hi   = lane >> 4;   // half-wave select

  __shared__ float ts[16 * VV]; // temporal sums per (row m, vertex v)
  __shared__ float scal[4];     // sum1, sumsq1, sum2, sumsq2

  for (int i = tid; i < 16 * VV; i += 256) ts[i] = 0.0f;
  if (tid < 4) scal[tid] = 0.0f;
  __syncthreads();

  const float* xb = x + (size_t)n * (CIN * PP);
  float s_loc = 0.0f, q_loc = 0.0f;

  // 200 position-tiles of 16; wave-strided (exactly 25 per wave)
  for (int pt = wave; pt < PP / 16; pt += 8) {
    // init C with biases: row m = i + 8*hi; rows 0-7 -> branch1, 8-15 -> branch2
    v8f c;
#pragma unroll
    for (int i = 0; i < 8; ++i) {
      const int m = i + hi * 8;
      c[i] = (m < 8) ? b1[m] : b2[m - 8];
    }
    const int pos = pt * 16 + lo;
#pragma unroll
    for (int kc = 0; kc < 16; ++kc) {       // K = 64 channels in chunks of 4
      const int cc = kc * 4 + hi * 2;
      v2f a, b;
      const int m = lo;
      a.x = (m < 8) ? w1[m * CIN + cc]     : w2[(m - 8) * CIN + cc];
      a.y = (m < 8) ? w1[m * CIN + cc + 1] : w2[(m - 8) * CIN + cc + 1];
      b.x = xb[(size_t)cc * PP + pos];
      b.y = xb[(size_t)(cc + 1) * PP + pos];
      c = wmma_f32(a, b, c);
    }
    const int v = pos % VV;
#pragma unroll
    for (int i = 0; i < 8; ++i) {
      const int m = i + hi * 8;
      const float val = c[i];
      s_loc += val;
      q_loc += val * val;
      atomicAdd(&ts[m * VV + v], val);
    }
  }
  // half-wave hi==0 holds only branch-1 rows, hi==1 only branch-2 rows
  if (hi == 0) { atomicAdd(&scal[0], s_loc); atomicAdd(&scal[1], q_loc); }
  else         { atomicAdd(&scal[2], s_loc); atomicAdd(&scal[3], q_loc); }
  __syncthreads();

  const float invN = 1.0f / (float)(RR * PP);
  const float mu1 = scal[0] * invN;
  const float va1 = scal[1] * invN - mu1 * mu1;
  const float mu2 = scal[2] * invN;
  const float va2 = scal[3] * invN - mu2 * mu2;
  const float rs1 = rsqrtf(va1 + EPSV);
  const float rs2 = rsqrtf(va2 + EPSV);

  for (int i = tid; i < 16 * VV; i += 256) {
    const int m = i / VV, v = i % VV;
    const float mt = ts[i] * (1.0f / (float)TT);   // mean over T
    if (m < 8)
      x1o[(n * RR + m) * VV + v] = g1[m] * ((mt - mu1) * rs1) + be1[m];
    else
      x2o[(n * RR + (m - 8)) * VV + v] = g2[m - 8] * ((mt - mu2) * rs2) + be2[m - 8];
  }
}

// ---------------------------------------------------------------------------
// Kernel 2: ysum[n,o,u,v] = sum_r w4[o,r]*(tanh(x1u-x1v)+tanh(x2u*x2v)+A[u,v]) + b4[o]
// grid = N, block = 256. Tiny VALU kernel; output (20.5 MB) stays in L2.
// ---------------------------------------------------------------------------
__global__ __launch_bounds__(256) void k_ysum(
    const float* __restrict__ x1, const float* __restrict__ x2,
    const float* __restrict__ Aadj,
    const float* __restrict__ w4, const float* __restrict__ b4,
    float* __restrict__ ysum) {
  const int n = blockIdx.x;
  const int tid = threadIdx.x;

  __shared__ float sx1[RR * VV], sx2[RR * VV], sw4[OO * RR], sb4[OO];
  for (int i = tid; i < RR * VV; i += 256) {
    sx1[i] = x1[n * RR * VV + i];
    sx2[i] = x2[n * RR * VV + i];
  }
  for (int i = tid; i < OO * RR; i += 256) sw4[i] = w4[i];
  if (tid < OO) sb4[tid] = b4[tid];
  __syncthreads();

  for (int p = tid; p < VV * VV; p += 256) {
    const int u = p / VV, v = p % VV;
    const float auv = Aadj[p];
    float s[RR];
#pragma unroll
    for (int r = 0; r < RR; ++r) {
      s[r] = tanhf(sx1[r * VV + u] - sx1[r * VV + v]) +
             tanhf(sx2[r * VV + u] * sx2[r * VV + v]) + auv;
    }
    float* yb = ysum + (size_t)n * OO * (VV * VV) + p;
#pragma unroll 4
    for (int o = 0; o < OO; ++o) {
      float acc = sb4[o];
#pragma unroll
      for (int r = 0; r < RR; ++r) acc += sw4[o * RR + r] * s[r];
      yb[(size_t)o * (VV * VV)] = acc;
    }
  }
}

// ---------------------------------------------------------------------------
// Kernel 3: fused  x3 = w3@x + b3  (WMMA GEMM1, K=64)  then
//           y[n,o,t,u] = sum_v x3[n,o,t,v] * ysum[n,o,u,v]  (WMMA GEMM2, K=25->28)
// grid = (T/16, N), block = 256 (8 waves). O handled in 4 quarters of 16 so
// LDS = xs[16][400] + x3q[16][400] = 51.2 KB.
// x tile staging uses the CDNA5 async global->LDS path (ASYNCcnt): VGPR-free
// b128 copies, drained (asynccnt 0) before the WMMA stream.
// ---------------------------------------------------------------------------
__global__ __launch_bounds__(256) void k_main(
    const float* __restrict__ x,
    const float* __restrict__ w3, const float* __restrict__ b3,
    const float* __restrict__ ysum,
    float* __restrict__ out) {
  const int tb   = blockIdx.x;          // t-tile (16 t values)
  const int n    = blockIdx.y;
  const int tid  = threadIdx.x;
  const int wave = tid >> 5;
  const int lane = tid & 31;
  const int lo   = lane & 15;
  const int hi   = lane >> 4;

  __shared__ float xs[16][400];    // 16-channel chunk of x tile [c][pos]
  __shared__ float x3q[16][400];   // x3 quarter [o_local][pos], pos = t_local*25+v

  const float* xg = x + (size_t)n * (CIN * PP) + (size_t)tb * 400;

  for (int q = 0; q < 4; ++q) {
    const int obase = q * 16;

    // --- GEMM1: x3[obase..obase+15][400 pos], K = 64 channels -------------
    v8f acc[4];
#pragma unroll
    for (int s = 0; s < 4; ++s) {
      v8f c;
#pragma unroll
      for (int i = 0; i < 8; ++i) c[i] = b3[obase + i + hi * 8];
      acc[s] = c;
    }

    for (int chunk = 0; chunk < 4; ++chunk) {
      __syncthreads();
      const int cb = chunk * 16;
#if USE_ASYNC_LDS
      // 16 rows x 400 floats = 1600 aligned 16-byte transfers, no VGPR bounce
      for (int idx = tid; idx < 1600; idx += 256) {
        const int cl = idx / 100, pq = idx % 100;   // row, 16B-chunk in row
        const float* gp = xg + (size_t)(cb + cl) * PP + pq * 4;
        __builtin_amdgcn_global_load_async_to_lds_b128(
            (gv4i_p)gp, (lv4i_p)&xs[cl][pq * 4],
            /*offset=*/0, /*cpol=*/0);
      }
      wait_asynccnt0();
#else
      for (int i = tid; i < 16 * 400; i += 256) {
        const int cl = i / 400, pos = i % 400;
        xs[cl][pos] = xg[(size_t)(cb + cl) * PP + pos];  // t,v contiguous
      }
#endif
      __syncthreads();
#pragma unroll
      for (int s = 0; s < 4; ++s) {
        const int pt = wave + s * 8;       // position tile 0..24 (wave-uniform)
        if (pt < 25) {
          const int npos = pt * 16 + lo;
#pragma unroll
          for (int kc = 0; kc < 4; ++kc) {
            const int cc = kc * 4 + hi * 2;
            v2f a, b;
            a.x = w3[(obase + lo) * CIN + cb + cc];
            a.y = w3[(obase + lo) * CIN + cb + cc + 1];
            b.x = xs[cc][npos];
            b.y = xs[cc + 1][npos];
            acc[s] = wmma_f32(a, b, acc[s]);
          }
        }
      }
    }

    // park x3 quarter in LDS: D row m = i + 8*hi -> o_local, col = pos
    __syncthreads();
#pragma unroll
    for (int s = 0; s < 4; ++s) {
      const int pt = wave + s * 8;
      if (pt < 25) {
#pragma unroll
        for (int i = 0; i < 8; ++i)
          x3q[i + hi * 8][pt * 16 + lo] = acc[s][i];
      }
    }
    __syncthreads();

    // --- GEMM2: y[t,u] = sum_v x3[t,v] * ysum[u,v] ------------------------
    // 16 o_local x 2 u-tiles = 32 jobs, 4 per wave. v padded 25->28.
#pragma unroll
    for (int jj = 0; jj < 4; ++jj) {
      const int j  = wave + jj * 8;       // 0..31 (wave-uniform)
      const int ol = j >> 1;
      const int ut = j & 1;
      const int o  = obase + ol;
      const float* ysb = ysum + ((size_t)n * OO + o) * (VV * VV);
      const int u = ut * 16 + lo;

      v8f d = {};
#pragma unroll
      for (int kk = 0; kk < 7; ++kk) {
        const int vx = kk * 4 + hi * 2;
        v2f a, b;
        a.x = (vx     < VV) ? x3q[ol][lo * VV + vx]     : 0.0f;
        a.y = (vx + 1 < VV) ? x3q[ol][lo * VV + vx + 1] : 0.0f;
        b.x = (u < VV && vx     < VV) ? ysb[u * VV + vx]     : 0.0f;
        b.y = (u < VV && vx + 1 < VV) ? ysb[u * VV + vx + 1] : 0.0f;
        d = wmma_f32(a, b, d);
      }
      if (u < VV) {
#pragma unroll
        for (int i = 0; i < 8; ++i) {
          const int t = tb * 16 + i + hi * 8;
          out[(((size_t)n * OO + o) * TT + t) * VV + u] = d[i];
        }
      }
    }
    __syncthreads();   // x3q / xs reused next quarter
  }
}

// ---------------------------------------------------------------------------
extern "C" void kernel_launch(void* const* d_in, const int* in_sizes, int n_in,
                              void* d_out, int out_size, void* d_ws, size_t ws_size,
                              hipStream_t stream) {
  const float* x   = (const float*)d_in[0];
  const float* A   = (const float*)d_in[1];
  const float* w1  = (const float*)d_in[2];
  const float* b1  = (const float*)d_in[3];
  const float* g1  = (const float*)d_in[4];
  const float* be1 = (const float*)d_in[5];
  const float* w2  = (const float*)d_in[6];
  const float* b2  = (const float*)d_in[7];
  const float* g2  = (const float*)d_in[8];
  const float* be2 = (const float*)d_in[9];
  const float* w3  = (const float*)d_in[10];
  const float* b3  = (const float*)d_in[11];
  const float* w4  = (const float*)d_in[12];
  const float* b4  = (const float*)d_in[13];
  float* out = (float*)d_out;

  // workspace layout (floats): x1 | x2 | ysum  (~20.7 MB total)
  float* ws   = (float*)d_ws;
  float* x1w  = ws;
  float* x2w  = ws + NB * RR * VV;
  float* ysum = ws + 2 * NB * RR * VV;

  k_stats<<<dim3(NB), dim3(256), 0, stream>>>(x, w1, b1, g1, be1,
                                              w2, b2, g2, be2, x1w, x2w);
  k_ysum<<<dim3(NB), dim3(256), 0, stream>>>(x1w, x2w, A, w4, b4, ysum);
  k_main<<<dim3(TT / 16, NB), dim3(256), 0, stream>>>(x, w3, b3, ysum, out);
}